// SemanticSemiSymbolic_21912923144501
// MI455X (gfx1250) — compile-verified
//
#include <hip/hip_runtime.h>
#include <math.h>

typedef float v2f __attribute__((ext_vector_type(2)));
typedef float v8f __attribute__((ext_vector_type(8)));

#define B_DIM   256
#define IN_DIM  512
#define OUT_DIM 512
#define E_DIM   256
#define DELTA_F 0.1f
#define LN_EPS_F 1e-5f

// ---------------------------------------------------------------------------
// Kernel 0: generic 32x32-tile transpose: src (R x C) -> dst (C x R).
// Used for weights (512x512) -> wT and attn_w (512x256) -> awT so all
// B-operand / elementwise streams become coalesced.
// ---------------------------------------------------------------------------
__global__ __launch_bounds__(1024)
void transpose_mat(const float* __restrict__ src, float* __restrict__ dst,
                   int R, int C) {
    __shared__ float tile[32][33];
    int gx = blockIdx.x;   // C tile
    int gy = blockIdx.y;   // R tile
    int tx = threadIdx.x, ty = threadIdx.y;
    tile[ty][tx] = src[(gy * 32 + ty) * C + gx * 32 + tx];
    __syncthreads();
    dst[(gx * 32 + ty) * R + gy * 32 + tx] = tile[tx][ty];
}

// ---------------------------------------------------------------------------
// Kernel 1: v[i,o] = |weights[o,i]| * sigmoid(LN(pred_emb @ attn_w^T + b))
// One block = 16 rows (i) x 512 cols (o): 32 waves, one 16x16 WMMA tile each
// (K=256 via v_wmma_f32_16x16x4_f32, two independent accumulator chains).
// A-operand staged in LDS (conflict-free), B-operand from transposed awT
// (coalesced). Then block-wide LayerNorm.
// ---------------------------------------------------------------------------
__global__ __launch_bounds__(1024)
void compute_v(const float* __restrict__ pe,  const float* __restrict__ awT,
               const float* __restrict__ ab,  const float* __restrict__ lng,
               const float* __restrict__ lnb, const float* __restrict__ wT,
               float* __restrict__ v) {
    __shared__ __align__(16) float s_lds[16][512];    // 32 KB scores tile
    __shared__ __align__(16) float pe_lds[E_DIM][16]; // 16 KB A tile (k-major)
    __shared__ float red1[16 * 64];
    __shared__ float red2[16 * 64];
    __shared__ float mu_s[16], rs_s[16];

    const int ibase = blockIdx.x * 16;
    const int tid   = threadIdx.x;

    // Stage pred_emb tile (coalesced per 64-thread group)
    {
        int m  = tid >> 6;                  // 0..15
        int ks = tid & 63;
        for (int c = ks; c < E_DIM; c += 64)
            pe_lds[c][m] = pe[(ibase + m) * E_DIM + c];
    }
    __syncthreads();

    const int wv    = tid >> 5;
    const int lane  = tid & 31;
    const int half  = lane >> 4;            // 0 or 1
    const int idx   = lane & 15;            // 0..15
    const int obase = wv * 16;
    const int kk    = half * 2;             // K-slot per ISA 16x4/4x16 layout

    v8f accA = {}, accB = {};               // two independent WMMA chains
    for (int k0 = 0; k0 < E_DIM; k0 += 8) {
        v2f a, b;
        a.x = pe_lds[k0 + kk][idx];
        a.y = pe_lds[k0 + kk + 1][idx];
        const float* pb = awT + (k0 + kk) * OUT_DIM + obase + idx;
        b.x = pb[0];
        b.y = pb[OUT_DIM];
        accA = __builtin_amdgcn_wmma_f32_16x16x4_f32(
            false, a, false, b, (short)0, accA, false, false);

        a.x = pe_lds[k0 + 4 + kk][idx];
        a.y = pe_lds[k0 + 4 + kk + 1][idx];
        const float* pb2 = awT + (k0 + 4 + kk) * OUT_DIM + obase + idx;
        b.x = pb2[0];
        b.y = pb2[OUT_DIM];
        accB = __builtin_amdgcn_wmma_f32_16x16x4_f32(
            false, a, false, b, (short)0, accB, false, false);
    }

    const float bias_o = ab[obase + idx];
    #pragma unroll
    for (int r = 0; r < 8; ++r)             // C layout: M = r + 8*half, N = idx
        s_lds[r + 8 * half][obase + idx] = accA[r] + accB[r] + bias_o;
    __syncthreads();

    // LayerNorm: per-row sum and sum-of-squares (64 threads per row)
    const int row = tid >> 6;               // 0..15
    const int j   = tid & 63;               // 0..63
    float s1 = 0.f, s2 = 0.f;
    for (int c = j; c < 512; c += 64) {
        float sv = s_lds[row][c];
        s1 += sv; s2 += sv * sv;
    }
    red1[row * 64 + j] = s1;
    red2[row * 64 + j] = s2;
    __syncthreads();
    for (int st = 32; st > 0; st >>= 1) {
        if (j < st) {
            red1[row * 64 + j] += red1[row * 64 + j + st];
            red2[row * 64 + j] += red2[row * 64 + j + st];
        }
        __syncthreads();
    }
    if (tid < 16) {
        float mu  = red1[tid * 64] * (1.0f / 512.0f);
        float var = red2[tid * 64] * (1.0f / 512.0f) - mu * mu;
        mu_s[tid] = mu;
        rs_s[tid] = rsqrtf(var + LN_EPS_F);
    }
    __syncthreads();

    // v[i,o] = |wT[i,o]| * sigmoid(norm * g[o] + b[o]); fully coalesced
    #pragma unroll
    for (int q = 0; q < 8; ++q) {
        int pos = q * 1024 + tid;
        int r   = pos >> 9;
        int c   = pos & 511;
        float norm = (s_lds[r][c] - mu_s[r]) * rs_s[r];
        float y    = norm * lng[c] + lnb[c];
        float sg   = 1.0f / (1.0f + __expf(-y));
        int   i    = ibase + r;
        v[i * OUT_DIM + c] = fabsf(wT[i * OUT_DIM + c]) * sg;
    }
}

// ---------------------------------------------------------------------------
// Kernel 2: out[b,o] = x@W^T + DELTA * (max_i u*v - u@v),  u = x*|x|
// Block: one 16x16 output tile; the 8 waves split the i (=K) dimension into
// 64-wide slices (4096 waves in flight), then combine partial sums (add)
// and partial maxes (max) through LDS. Four independent WMMA chains.
// ---------------------------------------------------------------------------
__global__ __launch_bounds__(256)
void fused_main(const float* __restrict__ x, const float* __restrict__ wT,
                const float* __restrict__ v, float* __restrict__ out) {
    __shared__ __align__(16) float x_lds[IN_DIM][16];   // 32 KB
    __shared__ __align__(16) float u_lds[IN_DIM][16];   // 32 KB
    __shared__ float r_acc1[8][256];        // 8 KB partial x@W^T
    __shared__ float r_acc2[8][256];        // 8 KB partial u@v
    __shared__ float r_mx  [8][256];        // 8 KB partial max

    const int bbase = blockIdx.y * 16;      // B tile
    const int obase = blockIdx.x * 16;      // OUT tile
    const int tid   = threadIdx.x;

    // Stage x tile and u = x*|x| (coalesced 64B chunks per 16-thread group)
    {
        int m  = tid >> 4;                  // 0..15 b-row
        int ks = tid & 15;
        for (int c = ks; c < IN_DIM; c += 16) {
            float xv = x[(bbase + m) * IN_DIM + c];
            x_lds[c][m] = xv;
            u_lds[c][m] = xv * fabsf(xv);
        }
    }
    __syncthreads();

    const int wv    = tid >> 5;
    const int lane  = tid & 31;
    const int half  = lane >> 4;
    const int idx   = lane & 15;
    const int kk    = half * 2;
    const int kbeg  = wv * (IN_DIM / 8);    // this wave's 64-wide K slice
    const int kend  = kbeg + (IN_DIM / 8);

    v8f acc1a = {}, acc1b = {};             // x @ W^T  (2 chains)
    v8f acc2a = {}, acc2b = {};             // u @ v    (2 chains)
    for (int k0 = kbeg; k0 < kend; k0 += 8) {
        v2f a1, a2, b1, b2;
        a1.x = x_lds[k0 + kk][idx];  a1.y = x_lds[k0 + kk + 1][idx];
        a2.x = u_lds[k0 + kk][idx];  a2.y = u_lds[k0 + kk + 1][idx];
        const float* p1 = wT + (k0 + kk) * OUT_DIM + obase + idx;
        b1.x = p1[0];  b1.y = p1[OUT_DIM];
        const float* p2 = v  + (k0 + kk) * OUT_DIM + obase + idx;
        b2.x = p2[0];  b2.y = p2[OUT_DIM];
        acc1a = __builtin_amdgcn_wmma_f32_16x16x4_f32(
            false, a1, false, b1, (short)0, acc1a, false, false);
        acc2a = __builtin_amdgcn_wmma_f32_16x16x4_f32(
            false, a2, false, b2, (short)0, acc2a, false, false);

        a1.x = x_lds[k0 + 4 + kk][idx];  a1.y = x_lds[k0 + 4 + kk + 1][idx];
        a2.x = u_lds[k0 + 4 + kk][idx];  a2.y = u_lds[k0 + 4 + kk + 1][idx];
        const float* p3 = wT + (k0 + 4 + kk) * OUT_DIM + obase + idx;
        b1.x = p3[0];  b1.y = p3[OUT_DIM];
        const float* p4 = v  + (k0 + 4 + kk) * OUT_DIM + obase + idx;
        b2.x = p4[0];  b2.y = p4[OUT_DIM];
        acc1b = __builtin_amdgcn_wmma_f32_16x16x4_f32(
            false, a1, false, b1, (short)0, acc1b, false, false);
        acc2b = __builtin_amdgcn_wmma_f32_16x16x4_f32(
            false, a2, false, b2, (short)0, acc2b, false, false);
    }

    // Max-product over this wave's i-slice (VALU; float4 LDS reads)
    float mx[8];
    #pragma unroll
    for (int r = 0; r < 8; ++r) mx[r] = -3.402823466e38f;
    #pragma unroll 4
    for (int i = kbeg; i < kend; ++i) {
        float  vv = v[i * OUT_DIM + obase + idx];          // coalesced, L2
        float4 u0 = *(const float4*)&u_lds[i][8 * half];   // ds_load_b128
        float4 u1 = *(const float4*)&u_lds[i][8 * half + 4];
        mx[0] = fmaxf(mx[0], u0.x * vv);
        mx[1] = fmaxf(mx[1], u0.y * vv);
        mx[2] = fmaxf(mx[2], u0.z * vv);
        mx[3] = fmaxf(mx[3], u0.w * vv);
        mx[4] = fmaxf(mx[4], u1.x * vv);
        mx[5] = fmaxf(mx[5], u1.y * vv);
        mx[6] = fmaxf(mx[6], u1.z * vv);
        mx[7] = fmaxf(mx[7], u1.w * vv);
    }

    // Publish partials (C layout: element M = r + 8*half, N = idx)
    #pragma unroll
    for (int r = 0; r < 8; ++r) {
        int el = (r + 8 * half) * 16 + idx;
        r_acc1[wv][el] = acc1a[r] + acc1b[r];
        r_acc2[wv][el] = acc2a[r] + acc2b[r];
        r_mx  [wv][el] = mx[r];
    }
    __syncthreads();

    // Cross-wave combine: thread tid owns output element (m = tid>>4, n = tid&15)
    float s1 = 0.f, s2 = 0.f, m_ = -3.402823466e38f;
    #pragma unroll
    for (int w = 0; w < 8; ++w) {
        s1 += r_acc1[w][tid];
        s2 += r_acc2[w][tid];
        m_  = fmaxf(m_, r_mx[w][tid]);
    }
    out[(bbase + (tid >> 4)) * OUT_DIM + obase + (tid & 15)] =
        s1 + DELTA_F * (m_ - s2);
}

// ---------------------------------------------------------------------------
extern "C" void kernel_launch(void* const* d_in, const int* in_sizes, int n_in,
                              void* d_out, int out_size, void* d_ws, size_t ws_size,
                              hipStream_t stream) {
    const float* x        = (const float*)d_in[0];
    const float* weights  = (const float*)d_in[1];
    const float* pred_emb = (const float*)d_in[2];
    const float* attn_w   = (const float*)d_in[3];
    const float* attn_b   = (const float*)d_in[4];
    const float* ln_g     = (const float*)d_in[5];
    const float* ln_b     = (const float*)d_in[6];
    float* out  = (float*)d_out;

    float* v_ws = (float*)d_ws;                    // IN*OUT   floats (1 MB)
    float* wT   = v_ws + IN_DIM * OUT_DIM;         // IN*OUT   floats (1 MB)
    float* awT  = wT   + IN_DIM * OUT_DIM;         // E*OUT    floats (0.5 MB)

    // weights (OUT x IN) -> wT (IN x OUT)
    transpose_mat<<<dim3(IN_DIM / 32, OUT_DIM / 32), dim3(32, 32), 0, stream>>>(
        weights, wT, OUT_DIM, IN_DIM);
    // attn_w (OUT x E) -> awT (E x OUT)
    transpose_mat<<<dim3(E_DIM / 32, OUT_DIM / 32), dim3(32, 32), 0, stream>>>(
        attn_w, awT, OUT_DIM, E_DIM);

    compute_v<<<IN_DIM / 16, 1024, 0, stream>>>(
        pred_emb, awT, attn_b, ln_g, ln_b, wT, v_ws);

    fused_main<<<dim3(OUT_DIM / 16, B_DIM / 16), 256, 0, stream>>>(
        x, wT, v_ws, out);
}